// LengthRegulator_29437705847511
// MI455X (gfx1250) — compile-verified
//
#include <hip/hip_runtime.h>

// LengthRegulator for MI455X (gfx1250).
// Pure gather/copy op: memory-bound, ~192MB HBM traffic -> ~8us floor @ 23.3TB/s.
// No matrix math => WMMA inapplicable; we use the CDNA5 async global<->LDS DMA
// path (global_load_async_to_lds_b128 / global_store_async_from_lds_b128,
// ASYNCcnt) as the row-copy engine: 16B per lane, zero VGPR data movement.

#define B_   32
#define P_   1024
#define D_   384
#define T_   3072

// ---------------- Kernel A: per-batch inclusive cumsum + lengths ------------
#define SCAN_TB 256
__global__ void lr_scan_kernel(const int* __restrict__ dur,
                               int* __restrict__ csum,
                               int* __restrict__ lens,
                               int* __restrict__ out_lens) {
    const int b   = blockIdx.x;
    const int tid = threadIdx.x;
    __shared__ int s_sum[SCAN_TB];

    const int4 v = ((const int4*)(dur + (size_t)b * P_))[tid];  // 4 consecutive durations
    const int l0 = v.x;
    const int l1 = l0 + v.y;
    const int l2 = l1 + v.z;
    const int l3 = l2 + v.w;

    s_sum[tid] = l3;
    __syncthreads();
    // Hillis-Steele inclusive scan over per-thread sums
    for (int off = 1; off < SCAN_TB; off <<= 1) {
        int val = (tid >= off) ? s_sum[tid - off] : 0;
        __syncthreads();
        s_sum[tid] += val;
        __syncthreads();
    }
    const int excl = (tid > 0) ? s_sum[tid - 1] : 0;

    int4 o;
    o.x = excl + l0; o.y = excl + l1; o.z = excl + l2; o.w = excl + l3;
    ((int4*)(csum + (size_t)b * P_))[tid] = o;

    if (tid == SCAN_TB - 1) {
        const int total = excl + l3;
        const int len   = total < T_ ? total : T_;
        lens[b]     = len;
        out_lens[b] = len;   // int bits into the flat output tail
    }
}

// ---------------- Kernel B: searchsorted + async row gather -----------------
#define ROWS   8                         // output rows (t values) per block
#define CPR    (D_ / 4)                  // 96 float4 chunks per row
#define NCHNK  (ROWS * CPR)              // 768 chunks per block
#define GTB    256                       // threads per block (8 waves)

__global__ void lr_gather_kernel(const float* __restrict__ x,
                                 const int*   __restrict__ csum,
                                 const int*   __restrict__ lens,
                                 float*       __restrict__ out) {
    __shared__ int    s_idx[ROWS];
    __shared__ int    s_valid[ROWS];
    __shared__ float4 s_buf[NCHNK];      // 12 KB staging buffer

    const int b   = blockIdx.y;
    const int t0  = blockIdx.x * ROWS;
    const int tid = threadIdx.x;

    const int* __restrict__ c = csum + (size_t)b * P_;
    const int len = lens[b];

    if (tid < ROWS) {
        const int t = t0 + tid;
        // searchsorted(csum, t, side='right'), then clip to P-1
        int lo = 0, hi = P_;
        while (lo < hi) {
            const int mid = (lo + hi) >> 1;
            if (c[mid] <= t) lo = mid + 1; else hi = mid;
        }
        if (lo > P_ - 1) lo = P_ - 1;
        s_idx[tid]   = lo;
        s_valid[tid] = (t < len) ? 1 : 0;
    }
    __syncthreads();

    const float* __restrict__ xb = x + (size_t)b * P_ * D_;
    float*       __restrict__ ob = out + ((size_t)b * T_ + (size_t)t0) * D_;

    // Phase 1: issue async global->LDS loads (16B/lane); masked rows get DS zeros.
    // Each thread owns chunks {tid, tid+256, tid+512} for BOTH load and store,
    // so per-wave s_wait_asynccnt/s_wait_dscnt suffice (no cross-thread barrier).
#pragma unroll
    for (int k = 0; k < NCHNK / GTB; ++k) {
        const int cidx = tid + k * GTB;
        const int r    = cidx / CPR;
        const int o    = cidx - r * CPR;
        const unsigned lds_off = (unsigned)(size_t)(&s_buf[cidx]);
        if (s_valid[r]) {
            const float4* src = (const float4*)(xb + (size_t)s_idx[r] * D_) + o;
            asm volatile("global_load_async_to_lds_b128 %0, %1, off"
                         :: "v"(lds_off),
                            "v"((unsigned long long)(size_t)src)
                         : "memory");
        } else {
            s_buf[cidx] = make_float4(0.f, 0.f, 0.f, 0.f);
        }
    }
    asm volatile("s_wait_asynccnt 0x0" ::: "memory");  // async loads landed in LDS
    asm volatile("s_wait_dscnt 0x0"    ::: "memory");  // DS zero-stores landed in LDS

    // Phase 2: async LDS->global stores of the staged rows.
#pragma unroll
    for (int k = 0; k < NCHNK / GTB; ++k) {
        const int cidx = tid + k * GTB;
        const int r    = cidx / CPR;
        const int o    = cidx - r * CPR;
        const unsigned lds_off = (unsigned)(size_t)(&s_buf[cidx]);
        float4* dst = (float4*)(ob + (size_t)r * D_) + o;
        asm volatile("global_store_async_from_lds_b128 %0, %1, off"
                     :: "v"((unsigned long long)(size_t)dst),
                        "v"(lds_off)
                     : "memory");
    }
    // S_ENDPGM performs an implicit wait-idle, draining outstanding async stores.
}

// ---------------------------------------------------------------------------
extern "C" void kernel_launch(void* const* d_in, const int* in_sizes, int n_in,
                              void* d_out, int out_size, void* d_ws, size_t ws_size,
                              hipStream_t stream) {
    const float* x   = (const float*)d_in[0];     // (B, P, D) f32
    const int*   dur = (const int*)d_in[1];       // (B, P) i32
    // d_in[2] = max_length scalar on device; grid must be host-known -> T_=3072.

    float* out      = (float*)d_out;                       // (B, T, D) f32
    int*   out_lens = (int*)((float*)d_out + (size_t)B_ * T_ * D_);  // (B,) i32 bits

    int* csum = (int*)d_ws;                // B*P ints = 128 KB
    int* lns  = csum + (size_t)B_ * P_;    // B ints

    lr_scan_kernel<<<B_, SCAN_TB, 0, stream>>>(dur, csum, lns, out_lens);
    lr_gather_kernel<<<dim3(T_ / ROWS, B_), GTB, 0, stream>>>(x, csum, lns, out);
}